// MH_U_MLP_11501922418779
// MI455X (gfx1250) — compile-verified
//
#include <hip/hip_runtime.h>
#include <hip/hip_bf16.h>
#include <math.h>

// ---------------------------------------------------------------------------
// MoE + linear3 for MI455X (gfx1250).
// Memory-bound (~0.6 GB traffic vs ~172 GFLOP fp32) -> keep fp32 end to end,
// use V_WMMA_F32_16X16X4_F32 for all GEMM work, VALU split-K for the 8x8 router.
// B tiles are k-paired (float2) in LDS: every WMMA fragment is one aligned
// ds_load_b64. All global->LDS staging is b128 loads + b128/b64 LDS stores.
// ---------------------------------------------------------------------------

#define B_ 8
#define S_ 2048
#define D_ 2048
#define E_ 8
#define TOPK_ 2
#define H_ 16
#define DH_ 128          // head dim == expert hidden dim
#define RK_ 4194304LL    // ROUTE_LEN * D = 2048*2048

typedef __attribute__((ext_vector_type(2))) float v2f;
typedef __attribute__((ext_vector_type(4))) float v4f;
typedef __attribute__((ext_vector_type(8))) float v8f;

// D = A(16x4) * B(4x16) + C, fp32, wave32.
__device__ __forceinline__ v8f wmma4(v2f a, v2f b, v8f c) {
  return __builtin_amdgcn_wmma_f32_16x16x4_f32(
      /*neg_a=*/false, a, /*neg_b=*/false, b,
      /*c_mod=*/(short)0, c, /*reuse_a=*/false, /*reuse_b=*/false);
}

// A fragment from row-major float tile viewed as v2f (pitch ldv in v2f units).
// lanes 0-15: M=lane, K pair (k0,k0+1); lanes 16-31: M=lane-16, K pair (k0+2,k0+3)
__device__ __forceinline__ v2f frag_a2(const v2f* As2, int ldv, int mbase, int k0, int lane) {
  int m  = mbase + (lane & 15);
  int kp = (k0 >> 1) + (lane >> 4);
  return As2[m * ldv + kp];
}

// B fragment from k-paired tile: Bsp[kp * ldn + n] = (B[2kp][n], B[2kp+1][n])
__device__ __forceinline__ v2f frag_b2(const v2f* Bsp, int ldn, int k0, int n0, int lane) {
  int n  = n0 + (lane & 15);
  int kp = (k0 >> 1) + (lane >> 4);
  return Bsp[kp * ldn + n];
}

// Stage a k-paired 32 x 128 B chunk: Bsp[k/2][n] = (W[n][k0g+k], W[n][k0g+k+1]).
// One b128 global load -> two b64 LDS stores per 4 elements; coalesced.
__device__ __forceinline__ void stage_b_paired(v2f* Bsp, const float* W, int ldw,
                                               int k0g, int tid) {
#pragma unroll
  for (int i = 0; i < 4; i++) {
    int lin = tid + i * 256;          // 0..1023 float4 slots
    int n   = lin >> 3;               // 0..127
    int kq  = (lin & 7) * 4;          // 0,4,...,28
    v4f v = *(const v4f*)&W[n * ldw + k0g + kq];
    v2f lo, hi;
    lo.x = v.x; lo.y = v.y;
    hi.x = v.z; hi.y = v.w;
    Bsp[(kq >> 1) * 144 + n]       = lo;
    Bsp[((kq >> 1) + 1) * 144 + n] = hi;
  }
}

#define BP_ 144   // v2f pitch of paired-B tiles: 2*144 mod 64 == 32 -> upper
                  // lanes (k+2 row) hit the opposite bank half: conflict-free

// ---------------------------------------------------------------------------
// Kernel 0: zero router logits accumulator (needed every call; atomics follow)
// ---------------------------------------------------------------------------
__global__ void init_logits(float* logits) {
  if (threadIdx.x < B_ * E_) logits[threadIdx.x] = 0.0f;
}

// ---------------------------------------------------------------------------
// Kernel 1: router partial GEMM  [8 x 4M] * [4M x 8], split-K over 512 chunks.
// ---------------------------------------------------------------------------
__global__ __launch_bounds__(256)
void router_partial(const float* __restrict__ x, const float* __restrict__ sw,
                    float* __restrict__ logits) {
  const long long chunk = RK_ / 512;            // 8192
  const long long kbeg  = (long long)blockIdx.x * chunk;
  float acc[B_][E_];
#pragma unroll
  for (int b = 0; b < B_; b++)
#pragma unroll
    for (int e = 0; e < E_; e++) acc[b][e] = 0.0f;

  for (long long k = kbeg + threadIdx.x; k < kbeg + chunk; k += 256) {
    float xv[B_], sv[E_];
#pragma unroll
    for (int b = 0; b < B_; b++) xv[b] = x[(long long)b * (S_ * (long long)D_) + k];
#pragma unroll
    for (int e = 0; e < E_; e++) sv[e] = sw[(long long)e * RK_ + k];
#pragma unroll
    for (int b = 0; b < B_; b++)
#pragma unroll
      for (int e = 0; e < E_; e++) acc[b][e] = fmaf(xv[b], sv[e], acc[b][e]);
  }

  // wave32 reduction
#pragma unroll
  for (int b = 0; b < B_; b++)
#pragma unroll
    for (int e = 0; e < E_; e++) {
      float v = acc[b][e];
      for (int off = 16; off > 0; off >>= 1) v += __shfl_down(v, off, 32);
      acc[b][e] = v;
    }

  __shared__ float part[8][B_ * E_];
  int lane = threadIdx.x & 31, wid = threadIdx.x >> 5;
  if (lane == 0) {
#pragma unroll
    for (int b = 0; b < B_; b++)
#pragma unroll
      for (int e = 0; e < E_; e++) part[wid][b * E_ + e] = acc[b][e];
  }
  __syncthreads();
  if (threadIdx.x < B_ * E_) {
    float s = 0.0f;
#pragma unroll
    for (int w = 0; w < 8; w++) s += part[w][threadIdx.x];
    atomicAdd(&logits[threadIdx.x], s);
  }
}

// ---------------------------------------------------------------------------
// Kernel 2: softmax + top-2 per batch sample (one lane per sample)
// ---------------------------------------------------------------------------
__global__ void router_topk(const float* __restrict__ logits,
                            const float* __restrict__ switch_b,
                            float* __restrict__ gate, int* __restrict__ tidx) {
  int b = threadIdx.x;
  if (b >= B_) return;
  float p[E_], mx = -1e30f;
#pragma unroll
  for (int e = 0; e < E_; e++) {
    p[e] = logits[b * E_ + e] + switch_b[e];
    mx = fmaxf(mx, p[e]);
  }
  float den = 0.0f;
#pragma unroll
  for (int e = 0; e < E_; e++) { p[e] = __expf(p[e] - mx); den += p[e]; }
  int i0 = 0; float v0 = p[0];
#pragma unroll
  for (int e = 1; e < E_; e++) if (p[e] > v0) { v0 = p[e]; i0 = e; }
  int i1 = (i0 == 0) ? 1 : 0; float v1 = p[i1];
#pragma unroll
  for (int e = 0; e < E_; e++) if (e != i0 && p[e] > v1) { v1 = p[e]; i1 = e; }
  gate[b * 2 + 0] = v0 / den;
  gate[b * 2 + 1] = v1 / den;
  tidx[b * 2 + 0] = i0;
  tidx[b * 2 + 1] = i1;
}

// ---------------------------------------------------------------------------
// Kernel 3: expert MLP (both top-k experts fused, gate-weighted) + residual.
// Block: 32 rows of one (b, h) slice.  8 waves, each owns a 16x32 tile.
// LDS: X tile (32x128), H tile (32x128), k-paired weight chunk (32x128).
// ---------------------------------------------------------------------------
__global__ __launch_bounds__(256)
void expert_mlp(const float* __restrict__ x,
                const float* __restrict__ w1, const float* __restrict__ b1,
                const float* __restrict__ w2, const float* __restrict__ b2,
                const float* __restrict__ gate, const int* __restrict__ tidx,
                float* __restrict__ y) {
  __shared__ __align__(16) float Xs[32 * 132];   // pitch 132 (mult of 4)
  __shared__ __align__(16) float Hs[32 * 132];
  __shared__ v2f Bsp[16 * BP_];                  // k-paired 32 x 128 chunk

  const int s0 = blockIdx.x * 32;
  const int h  = blockIdx.y;
  const int b  = blockIdx.z;
  const int tid = threadIdx.x, lane = tid & 31, wid = tid >> 5;
  const int rg = wid >> 2;       // row group: 0..1  (16 rows each)
  const int cg = wid & 3;        // col group: 0..3  (32 cols each)

  const float* xbase = x + (long long)b * S_ * D_ + (long long)s0 * D_ + h * DH_;
  float*       ybase = y + (long long)b * S_ * D_ + (long long)s0 * D_ + h * DH_;

  // stage X tile: 32 x 128 as float4 (b128 global load -> b128 LDS store)
#pragma unroll
  for (int i = 0; i < 4; i++) {
    int lin = tid + i * 256;          // float4 slot
    int r = lin >> 5;                 // 32 float4 per row
    int c = (lin & 31) * 4;
    *(v4f*)&Xs[r * 132 + c] = *(const v4f*)&xbase[(long long)r * D_ + c];
  }

  v8f accy0 = {}, accy1 = {};

  for (int kk = 0; kk < TOPK_; kk++) {
    const int   e = tidx[b * 2 + kk];
    const float g = gate[b * 2 + kk];
    const float* w1e = w1 + (long long)e * DH_ * DH_;
    const float* w2e = w2 + (long long)e * DH_ * DH_;
    const float* b1e = b1 + e * DH_;
    const float* b2e = b2 + e * DH_;

    // ---- GEMM1: H[s,f] = sum_c X[s,c] * w1[f,c] ----
    v8f h0 = {}, h1 = {};
    for (int kc = 0; kc < 4; kc++) {
      __syncthreads();
      stage_b_paired(Bsp, w1e, DH_, kc * 32, tid);
      __syncthreads();
#pragma unroll
      for (int k0 = 0; k0 < 32; k0 += 4) {
        v2f a  = frag_a2((const v2f*)Xs, 66, rg * 16, kc * 32 + k0, lane);
        v2f bA = frag_b2(Bsp, BP_, k0, cg * 32, lane);
        v2f bB = frag_b2(Bsp, BP_, k0, cg * 32 + 16, lane);
        h0 = wmma4(a, bA, h0);
        h1 = wmma4(a, bB, h1);
      }
    }

    // bias + exact GELU, write H tile to LDS
    const int nlo  = lane & 15;
    const int mofs = (lane >> 4) * 8;
#pragma unroll
    for (int r = 0; r < 8; r++) {
      int m = rg * 16 + r + mofs;
      float v0 = h0[r] + b1e[cg * 32 + nlo];
      float v1 = h1[r] + b1e[cg * 32 + 16 + nlo];
      v0 = 0.5f * v0 * (1.0f + erff(v0 * 0.70710678118654752f));
      v1 = 0.5f * v1 * (1.0f + erff(v1 * 0.70710678118654752f));
      Hs[m * 132 + cg * 32 + nlo]      = v0;
      Hs[m * 132 + cg * 32 + 16 + nlo] = v1;
    }

    // ---- GEMM2: O[s,d] = sum_f H[s,f] * w2[d,f] ----
    v8f o0 = {}, o1 = {};
    for (int kc = 0; kc < 4; kc++) {
      __syncthreads();                             // covers Hs writes on kc==0
      stage_b_paired(Bsp, w2e, DH_, kc * 32, tid);
      __syncthreads();
#pragma unroll
      for (int k0 = 0; k0 < 32; k0 += 4) {
        v2f a  = frag_a2((const v2f*)Hs, 66, rg * 16, kc * 32 + k0, lane);
        v2f bA = frag_b2(Bsp, BP_, k0, cg * 32, lane);
        v2f bB = frag_b2(Bsp, BP_, k0, cg * 32 + 16, lane);
        o0 = wmma4(a, bA, o0);
        o1 = wmma4(a, bB, o1);
      }
    }

    // gate-weighted accumulate (+ expert bias2)
#pragma unroll
    for (int r = 0; r < 8; r++) {
      accy0[r] += g * (o0[r] + b2e[cg * 32 + nlo]);
      accy1[r] += g * (o1[r] + b2e[cg * 32 + 16 + nlo]);
    }
  }

  // residual add + store y tile
  const int nlo  = lane & 15;
  const int mofs = (lane >> 4) * 8;
#pragma unroll
  for (int r = 0; r < 8; r++) {
    int m = rg * 16 + r + mofs;
    long long ro = (long long)m * D_;
    int c0 = cg * 32 + nlo;
    ybase[ro + c0]      = xbase[ro + c0]      + accy0[r];
    ybase[ro + c0 + 16] = xbase[ro + c0 + 16] + accy1[r];
  }
}

// ---------------------------------------------------------------------------
// Kernel 4: out = y @ w3^T + b3   ([16384,2048] x [2048,2048])
// 64x128 block tile, Kc=32 staging; 8 waves x (16x64), 4 accumulators each:
// one A fragment (ds_load_b64) feeds 4 WMMAs.
// ---------------------------------------------------------------------------
__global__ __launch_bounds__(256)
void linear3(const float* __restrict__ y, const float* __restrict__ w3,
             const float* __restrict__ b3, float* __restrict__ out) {
  __shared__ __align__(16) float As[64 * 36];    // pitch 36: b128-aligned rows
  __shared__ v2f Bsp[16 * BP_];                  // k-paired 32 x 128 chunk

  const int n0g = blockIdx.x * 128;
  const int m0  = blockIdx.y * 64;
  const int tid = threadIdx.x, lane = tid & 31, wid = tid >> 5;
  const int rg = wid >> 1;       // 0..3 (16 rows)
  const int cg = wid & 1;        // 0..1 (64 cols)

  v8f acc0 = {}, acc1 = {}, acc2 = {}, acc3 = {};

  for (int kc = 0; kc < 64; kc++) {
    __syncthreads();
#pragma unroll
    for (int i = 0; i < 2; i++) {                  // A tile 64 x 32 as float4
      int lin = tid + i * 256;                     // 0..511 float4 slots
      int r  = lin >> 3;                           // 8 float4 per row
      int kq = (lin & 7) * 4;
      *(v4f*)&As[r * 36 + kq] = *(const v4f*)&y[(long long)(m0 + r) * D_ + kc * 32 + kq];
    }
    stage_b_paired(Bsp, w3 + (long long)n0g * D_, D_, kc * 32, tid);
    if (kc < 63) {                                 // global_prefetch_b8 next A chunk
      __builtin_prefetch(&y[(long long)(m0 + wid * 8) * D_ + (kc + 1) * 32 + lane], 0, 1);
    }
    __syncthreads();
#pragma unroll
    for (int k0 = 0; k0 < 32; k0 += 4) {
      v2f a  = frag_a2((const v2f*)As, 18, rg * 16, k0, lane);
      v2f b0 = frag_b2(Bsp, BP_, k0, cg * 64, lane);
      v2f b1 = frag_b2(Bsp, BP_, k0, cg * 64 + 16, lane);
      v2f b2 = frag_b2(Bsp, BP_, k0, cg * 64 + 32, lane);
      v2f b3f = frag_b2(Bsp, BP_, k0, cg * 64 + 48, lane);
      acc0 = wmma4(a, b0, acc0);
      acc1 = wmma4(a, b1, acc1);
      acc2 = wmma4(a, b2, acc2);
      acc3 = wmma4(a, b3f, acc3);
    }
  }

  const int mofs = (lane >> 4) * 8;
  const int nb   = n0g + cg * 64 + (lane & 15);
#pragma unroll
  for (int r = 0; r < 8; r++) {
    int m = m0 + rg * 16 + r + mofs;
    long long ro = (long long)m * D_;
    out[ro + nb]      = acc0[r] + b3[nb];
    out[ro + nb + 16] = acc1[r] + b3[nb + 16];
    out[ro + nb + 32] = acc2[r] + b3[nb + 32];
    out[ro + nb + 48] = acc3[r] + b3[nb + 48];
  }
}

// ---------------------------------------------------------------------------
extern "C" void kernel_launch(void* const* d_in, const int* in_sizes, int n_in,
                              void* d_out, int out_size, void* d_ws, size_t ws_size,
                              hipStream_t stream) {
  const float* x        = (const float*)d_in[0];
  const float* switch_w = (const float*)d_in[1];
  const float* switch_b = (const float*)d_in[2];
  const float* w1       = (const float*)d_in[3];
  const float* b1       = (const float*)d_in[4];
  const float* w2       = (const float*)d_in[5];
  const float* b2       = (const float*)d_in[6];
  const float* w3       = (const float*)d_in[7];
  const float* b3       = (const float*)d_in[8];
  float* out = (float*)d_out;

  float* ws     = (float*)d_ws;
  float* logits = ws;                 // 64 floats
  float* gate   = ws + 64;            // 16 floats
  int*   tidx   = (int*)(ws + 80);    // 16 ints
  float* ybuf   = ws + 256;           // B*S*D floats (MoE output + residual)

  init_logits<<<1, 64, 0, stream>>>(logits);
  router_partial<<<512, 256, 0, stream>>>(x, switch_w, logits);
  router_topk<<<1, 32, 0, stream>>>(logits, switch_b, gate, tidx);
  expert_mlp<<<dim3(S_ / 32, H_, B_), 256, 0, stream>>>(x, w1, b1, w2, b2, gate, tidx, ybuf);
  linear3<<<dim3(D_ / 128, (B_ * S_) / 64), 256, 0, stream>>>(ybuf, w3, b3, out);
}